// GraphConv_70643622084871
// MI455X (gfx1250) — compile-verified
//
#include <hip/hip_runtime.h>
#include <hip/hip_bf16.h>

// ---------------------------------------------------------------------------
// GraphConv: out = (segment_sum over rows of vals[e]*X[col[e]]) @ W^T + bias
//   N = 100000 nodes, E = 1.6M edges, d_in = d_out = 64, all f32.
// Phase 1: zero S (workspace)        — S = 25.6 MB, fits in L2 (192 MB)
// Phase 2: scatter-add via f32 atomics (one wave per edge, float2 per lane)
// Phase 3: GEMM with V_WMMA_F32_16X16X4_F32 (full-precision f32 matrix core)
// ---------------------------------------------------------------------------

typedef __attribute__((ext_vector_type(2))) float v2f;
typedef __attribute__((ext_vector_type(8))) float v8f;

#define D_FEAT 64

// ---------------- Phase 1: zero the S accumulator ----------------
__global__ void gc_zero_kernel(float4* __restrict__ s4, int n4) {
    int i = blockIdx.x * blockDim.x + threadIdx.x;
    if (i < n4) {
        s4[i] = make_float4(0.f, 0.f, 0.f, 0.f);
    }
}

// ---------------- Phase 2: SpMM scatter-add ----------------
// One wave (32 lanes) per edge. row/col/val indices are wave-uniform
// (edge id depends only on blockIdx and wave id) -> scalar loads.
// Each lane handles 2 features: coalesced float2 gather from X, then two
// global_atomic_add_f32 into S. S stays resident in L2.
__global__ void gc_scatter_kernel(const int* __restrict__ row,
                                  const int* __restrict__ col,
                                  const float* __restrict__ vals,
                                  const float* __restrict__ X,
                                  float* __restrict__ S,
                                  int n_edges) {
    const int wave_in_block = threadIdx.x >> 5;       // 0..7 (256 threads = 8 waves)
    const int lane          = threadIdx.x & 31;
    const int e = blockIdx.x * 8 + wave_in_block;     // wave-uniform edge id
    if (e >= n_edges) return;

    const int   r = row[e];
    const int   c = col[e];
    const float v = vals[e];

    const float2 x = *(const float2*)(X + (size_t)c * D_FEAT + 2 * lane);
    float* dst = S + (size_t)r * D_FEAT + 2 * lane;
    atomicAdd(dst + 0, v * x.x);
    atomicAdd(dst + 1, v * x.y);
}

// ---------------- Phase 3: GEMM out = S @ W^T + bias via f32 WMMA ----------
// Block = 128 threads = 4 waves; block computes one 16-row M-tile.
// Wave w computes the 16x16 output tile for columns [16w, 16w+16).
// K = 64 accumulated as 16 steps of V_WMMA_F32_16X16X4_F32.
//
// A (16x4 f32, 2 VGPRs): lanes 0-15 hold row M=lane, K = 4k+{0,1};
//                        lanes 16-31 hold row M=lane-16, K = 4k+{2,3}.
// B (4x16 f32, 2 VGPRs): mirrored K split across lane halves; B[kk][o] = W[o][kk].
// C/D (16x16 f32, 8 VGPRs): VGPR r: lanes 0-15 -> M=r, lanes 16-31 -> M=r+8.
__global__ void gc_gemm_wmma_kernel(const float* __restrict__ S,
                                    const float* __restrict__ W,     // [64 x 64], row o = output feature
                                    const float* __restrict__ bias,  // [64]
                                    float* __restrict__ out,         // [N x 64]
                                    int n_nodes) {
    const int wave   = threadIdx.x >> 5;   // 0..3 -> output column tile
    const int lane   = threadIdx.x & 31;
    const int half   = lane >> 4;          // 0: lanes 0-15, 1: lanes 16-31
    const int l      = lane & 15;
    const int m_base = blockIdx.x * 16;
    if (m_base >= n_nodes) return;

    const int o = wave * 16 + l;           // this lane's output column (for B and D)

    const float* __restrict__ Srow = S + (size_t)(m_base + l) * D_FEAT;
    const float* __restrict__ Wrow = W + (size_t)o * D_FEAT;

    v8f acc = {};
#pragma unroll
    for (int k = 0; k < 16; ++k) {
        const int kk = 4 * k + 2 * half;   // this lane-half's K pair
        v2f a;
        a.x = Srow[kk + 0];
        a.y = Srow[kk + 1];
        v2f b;
        b.x = Wrow[kk + 0];
        b.y = Wrow[kk + 1];
        // (neg_a, A, neg_b, B, c_mod, C, reuse_a, reuse_b)
        acc = __builtin_amdgcn_wmma_f32_16x16x4_f32(
            false, a, false, b, (short)0, acc, false, false);
    }

    const float bo = bias[o];
#pragma unroll
    for (int r = 0; r < 8; ++r) {
        const int m = m_base + r + half * 8;
        out[(size_t)m * D_FEAT + o] = acc[r] + bo;
    }
}

// ---------------------------------------------------------------------------
extern "C" void kernel_launch(void* const* d_in, const int* in_sizes, int n_in,
                              void* d_out, int out_size, void* d_ws, size_t ws_size,
                              hipStream_t stream) {
    const int*   row  = (const int*)  d_in[0];
    const int*   col  = (const int*)  d_in[1];
    const float* vals = (const float*)d_in[2];
    const float* X    = (const float*)d_in[3];
    const float* Wt   = (const float*)d_in[4];   // weight [64, 64]
    const float* bias = (const float*)d_in[5];
    float*       out  = (float*)d_out;

    const int n_edges = in_sizes[0];
    const int n_nodes = out_size / D_FEAT;       // 100000

    float* S = (float*)d_ws;                     // [n_nodes x 64] f32 accumulator

    // Phase 1: zero S (must run every call; graph replays don't re-poison)
    {
        const int n4 = n_nodes * D_FEAT / 4;
        const int threads = 256;
        const int blocks = (n4 + threads - 1) / threads;
        gc_zero_kernel<<<blocks, threads, 0, stream>>>((float4*)S, n4);
    }

    // Phase 2: edge scatter (8 waves per block, one wave per edge)
    {
        const int threads = 256;
        const int blocks = (n_edges + 7) / 8;
        gc_scatter_kernel<<<blocks, threads, 0, stream>>>(row, col, vals, X, S, n_edges);
    }

    // Phase 3: WMMA GEMM + bias
    {
        const int threads = 128;                 // 4 waves
        const int blocks = (n_nodes + 15) / 16;  // 6250 M-tiles
        gc_gemm_wmma_kernel<<<blocks, threads, 0, stream>>>(S, Wt, bias, out, n_nodes);
    }
}